// GATRepresentationNetwork_72971494359376
// MI455X (gfx1250) — compile-verified
//
#include <hip/hip_runtime.h>
#include <hip/hip_bf16.h>

// CDNA5 (gfx1250) wave32 WMMA f32 16x16x4.
typedef __attribute__((ext_vector_type(2))) float v2f;
typedef __attribute__((ext_vector_type(8))) float v8f;

static __device__ __forceinline__ v8f wzero() {
  v8f v;
#pragma unroll
  for (int i = 0; i < 8; ++i) v[i] = 0.0f;
  return v;
}

// D = A(16x4 f32) * B(4x16 f32) + C(16x16 f32)
static __device__ __forceinline__ v8f wmma4(v2f a, v2f b, v8f c) {
  return __builtin_amdgcn_wmma_f32_16x16x4_f32(false, a, false, b, (short)0, c,
                                               false, false);
}

// ---------------------------------------------------------------------------
// Per-wave GAT layer (force-inlined so LDS pointers resolve to ds_* ops and
// FIN/MODE are compile-time):
//   MODE 0: hin <- relu(GAT(hin) + bias)       (concat heads, 256 wide)
//   MODE 2: grrow[64] <- mean over heads+nodes of GAT(hin) + bias
// x = hin @ W via WMMA; attention matrices P[h] (16x16) built from scores;
// aggregation agg = P_h @ x also via WMMA.
// ---------------------------------------------------------------------------
template <int FIN, int MODE>
static __device__ __forceinline__ void gat_layer(
    float* __restrict__ hin,    // LDS [16][256]
    float* __restrict__ xb,     // LDS [16][256] scratch
    float* __restrict__ Pb,     // LDS [4][16][16] scratch
    float* __restrict__ aab,    // LDS [128]: a_s[64], a_d[64]
    float* __restrict__ grrow,  // LDS [64] (MODE 2 only)
    const float* __restrict__ W,     // [FIN][256]
    const float* __restrict__ asrc,  // [4][64]
    const float* __restrict__ adst,  // [4][64]
    const float* __restrict__ bias,  // [256] (MODE 0) / [64] (MODE 2)
    int lane)
{
  const int lo = lane & 15;
  const int hi = lane >> 4;

  // (a) x = hin @ W  ->  xb   (16 column tiles, K-steps of 4)
  for (int t = 0; t < 16; ++t) {
    v8f acc = wzero();
#pragma unroll 4
    for (int k0 = 0; k0 < FIN; k0 += 4) {
      const int ka = k0 + 2 * hi;
      v2f a, b;
      a.x = hin[lo * 256 + ka];
      a.y = hin[lo * 256 + ka + 1];
      b.x = W[ka * 256 + t * 16 + lo];
      b.y = W[(ka + 1) * 256 + t * 16 + lo];
      acc = wmma4(a, b, acc);
    }
#pragma unroll
    for (int r = 0; r < 8; ++r)
      xb[(r + 8 * hi) * 256 + t * 16 + lo] = acc[r];
  }

  // (b) attention scores a_s/a_d per (head h, node n); p = h*16+n
#pragma unroll
  for (int q = 0; q < 2; ++q) {
    const int p = lane + q * 32;
    const int h = p >> 4;
    const int n = p & 15;
    float s = 0.f, d = 0.f;
#pragma unroll 8
    for (int c = 0; c < 64; ++c) {
      const float xv = xb[n * 256 + h * 64 + c];
      s += xv * asrc[h * 64 + c];
      d += xv * adst[h * 64 + c];
    }
    aab[p] = s;
    aab[64 + p] = d;
  }

  // (c) P[h][dst][src]: softmax over incoming edges (self + 4-neighborhood)
#pragma unroll
  for (int hb = 0; hb < 4; hb += 2) {
    const int h = hb + hi;   // two heads per pass across the two lane halves
    const int dst = lo;
    const float adv = aab[64 + h * 16 + dst];
    float al[16];
    float mx = -3.0e38f;
#pragma unroll
    for (int s = 0; s < 16; ++s) {
      const int di = (dst >> 2) - (s >> 2);
      const int dj = (dst & 3) - (s & 3);
      const int man = (di < 0 ? -di : di) + (dj < 0 ? -dj : dj);
      float a = aab[h * 16 + s] + adv;
      a = a > 0.f ? a : 0.2f * a;           // leaky relu, slope 0.2
      al[s] = (man <= 1) ? a : -3.0e38f;    // man==0 self loop, man==1 edge
      mx = fmaxf(mx, al[s]);
    }
    float sum = 0.f;
    float ex[16];
#pragma unroll
    for (int s = 0; s < 16; ++s) {
      const float e = (al[s] > -1.0e38f) ? __expf(al[s] - mx) : 0.f;
      ex[s] = e;
      sum += e;
    }
    const float inv = 1.f / sum;
#pragma unroll
    for (int s = 0; s < 16; ++s)
      Pb[h * 256 + dst * 16 + s] = ex[s] * inv;
  }

  // (d) aggregation agg = P_h @ x via WMMA
  if (MODE < 2) {
    for (int h = 0; h < 4; ++h) {
#pragma unroll
      for (int t = 0; t < 4; ++t) {
        v8f acc = wzero();
#pragma unroll
        for (int k0 = 0; k0 < 16; k0 += 4) {
          const int ka = k0 + 2 * hi;
          const int col = h * 64 + t * 16 + lo;
          v2f a, b;
          a.x = Pb[h * 256 + lo * 16 + ka];
          a.y = Pb[h * 256 + lo * 16 + ka + 1];
          b.x = xb[ka * 256 + col];
          b.y = xb[(ka + 1) * 256 + col];
          acc = wmma4(a, b, acc);
        }
        const int col = h * 64 + t * 16 + lo;
        const float bv = bias[col];
#pragma unroll
        for (int r = 0; r < 8; ++r) {
          const float v = acc[r] + bv;
          hin[(r + 8 * hi) * 256 + col] = v > 0.f ? v : 0.f;  // concat + relu
        }
      }
    }
  } else {
    // mean over heads, then mean over the 16 nodes, plus bias -> grrow[64]
#pragma unroll
    for (int t = 0; t < 4; ++t) {
      v8f acc = wzero();
#pragma unroll
      for (int h = 0; h < 4; ++h) {
#pragma unroll
        for (int k0 = 0; k0 < 16; k0 += 4) {
          const int ka = k0 + 2 * hi;
          const int col = h * 64 + t * 16 + lo;
          v2f a, b;
          a.x = Pb[h * 256 + lo * 16 + ka];
          a.y = Pb[h * 256 + lo * 16 + ka + 1];
          b.x = xb[ka * 256 + col];
          b.y = xb[(ka + 1) * 256 + col];
          acc = wmma4(a, b, acc);
        }
      }
      float s = 0.f;
#pragma unroll
      for (int r = 0; r < 8; ++r) s += acc[r];   // 8 node rows in this half
      s += __shfl_xor(s, 16);                    // + other 8 rows
      grrow[t * 16 + lo] = s * (1.0f / 64.0f) + bias[t * 16 + lo];
    }
  }
}

// ---------------------------------------------------------------------------
// Fully fused kernel: 1 workgroup = 8 waves = 8 batch elements.
// Per-wave LDS block of 9344 floats: hin(4096) | xb(4096) | Pb(1024) | aa(128)
// grbuf: [16][64] shared; MLP reuses low LDS after a barrier.
// ---------------------------------------------------------------------------
__global__ __launch_bounds__(256) void gat_fused(
    const float* __restrict__ x,
    const float* __restrict__ w_in, const float* __restrict__ b_in,
    const float* __restrict__ w0, const float* __restrict__ as0,
    const float* __restrict__ ad0, const float* __restrict__ bias0,
    const float* __restrict__ w1, const float* __restrict__ as1,
    const float* __restrict__ ad1, const float* __restrict__ bias1,
    const float* __restrict__ w2, const float* __restrict__ as2,
    const float* __restrict__ ad2, const float* __restrict__ bias2,
    const float* __restrict__ mw1, const float* __restrict__ mb1,
    const float* __restrict__ g1, const float* __restrict__ be1,
    const float* __restrict__ mw2, const float* __restrict__ mb2,
    const float* __restrict__ g2, const float* __restrict__ be2,
    float* __restrict__ out)
{
  extern __shared__ float smem[];
  const int lane = threadIdx.x & 31;
  const int wave = threadIdx.x >> 5;
  const int lo = lane & 15;
  const int hi = lane >> 4;
  const int b = blockIdx.x * 8 + wave;

  float* hin = smem + wave * 9344;
  float* xb = hin + 4096;
  float* Pb = hin + 8192;
  float* aab = hin + 9216;
  float* grbuf = smem + 8 * 9344;  // [16][64]

  // ---- input projection: h0 = relu(nodes(16) x cin(16) @ w_in + b_in) ----
  const float* xg = x + (size_t)b * 256;  // [cin=16][node=16]
#pragma unroll
  for (int t = 0; t < 4; ++t) {
    v8f acc = wzero();
#pragma unroll
    for (int k0 = 0; k0 < 16; k0 += 4) {
      const int ka = k0 + 2 * hi;
      v2f a, bv;
      a.x = xg[ka * 16 + lo];               // A[m=node][k=cin]
      a.y = xg[(ka + 1) * 16 + lo];
      bv.x = w_in[ka * 64 + t * 16 + lo];   // B[k=cin][n]
      bv.y = w_in[(ka + 1) * 64 + t * 16 + lo];
      acc = wmma4(a, bv, acc);
    }
    const float bb = b_in[t * 16 + lo];
#pragma unroll
    for (int r = 0; r < 8; ++r) {
      const float v = acc[r] + bb;
      hin[(r + 8 * hi) * 256 + t * 16 + lo] = v > 0.f ? v : 0.f;
    }
  }

  // ---- three GAT layers ----
  gat_layer<64, 0>(hin, xb, Pb, aab, nullptr, w0, as0, ad0, bias0, lane);
  gat_layer<256, 0>(hin, xb, Pb, aab, nullptr, w1, as1, ad1, bias1, lane);
  gat_layer<256, 2>(hin, xb, Pb, aab, grbuf + wave * 64, w2, as2, ad2, bias2,
                    lane);

  // zero-pad gr rows 8..15 so the MLP WMMA tile is well defined
  grbuf[(8 + wave) * 64 + lane] = 0.f;
  grbuf[(8 + wave) * 64 + 32 + lane] = 0.f;
  __syncthreads();

  // ---- MLP over the 8 pooled rows of this block (WMMA, M=16 padded) ----
  float* yb1 = smem;         // [16][128]
  float* yb2 = smem + 2048;  // [16][256]

  // GEMM1: y1 = gr(16x64) @ mw1(64x128) + mb1; tile t == wave
  {
    v8f acc = wzero();
#pragma unroll 4
    for (int k0 = 0; k0 < 64; k0 += 4) {
      const int ka = k0 + 2 * hi;
      v2f a, bv;
      a.x = grbuf[lo * 64 + ka];
      a.y = grbuf[lo * 64 + ka + 1];
      bv.x = mw1[ka * 128 + wave * 16 + lo];
      bv.y = mw1[(ka + 1) * 128 + wave * 16 + lo];
      acc = wmma4(a, bv, acc);
    }
    const float bb = mb1[wave * 16 + lo];
#pragma unroll
    for (int r = 0; r < 8; ++r)
      yb1[(r + 8 * hi) * 128 + wave * 16 + lo] = acc[r] + bb;
  }
  __syncthreads();

  // LN1 + relu on row `wave` (rows 0..7 valid)
  {
    float* row = yb1 + wave * 128;
    float s = 0.f;
#pragma unroll
    for (int c = lane; c < 128; c += 32) s += row[c];
#pragma unroll
    for (int m = 16; m >= 1; m >>= 1) s += __shfl_xor(s, m);
    const float mu = s * (1.0f / 128.0f);
    float vv = 0.f;
#pragma unroll
    for (int c = lane; c < 128; c += 32) {
      const float d = row[c] - mu;
      vv += d * d;
    }
#pragma unroll
    for (int m = 16; m >= 1; m >>= 1) vv += __shfl_xor(vv, m);
    const float inv = rsqrtf(vv * (1.0f / 128.0f) + 1e-5f);
#pragma unroll
    for (int c = lane; c < 128; c += 32) {
      const float val = (row[c] - mu) * inv * g1[c] + be1[c];
      row[c] = val > 0.f ? val : 0.f;
    }
  }
  __syncthreads();

  // GEMM2: y2 = y1(16x128) @ mw2(128x256) + mb2; tiles wave, wave+8
#pragma unroll
  for (int tt = 0; tt < 2; ++tt) {
    const int t = wave + tt * 8;
    v8f acc = wzero();
#pragma unroll 4
    for (int k0 = 0; k0 < 128; k0 += 4) {
      const int ka = k0 + 2 * hi;
      v2f a, bv;
      a.x = yb1[lo * 128 + ka];
      a.y = yb1[lo * 128 + ka + 1];
      bv.x = mw2[ka * 256 + t * 16 + lo];
      bv.y = mw2[(ka + 1) * 256 + t * 16 + lo];
      acc = wmma4(a, bv, acc);
    }
    const float bb = mb2[t * 16 + lo];
#pragma unroll
    for (int r = 0; r < 8; ++r)
      yb2[(r + 8 * hi) * 256 + t * 16 + lo] = acc[r] + bb;
  }
  __syncthreads();

  // LN2 + relu -> out row b
  {
    float* row = yb2 + wave * 256;
    float s = 0.f;
#pragma unroll
    for (int c = lane; c < 256; c += 32) s += row[c];
#pragma unroll
    for (int m = 16; m >= 1; m >>= 1) s += __shfl_xor(s, m);
    const float mu = s * (1.0f / 256.0f);
    float vv = 0.f;
#pragma unroll
    for (int c = lane; c < 256; c += 32) {
      const float d = row[c] - mu;
      vv += d * d;
    }
#pragma unroll
    for (int m = 16; m >= 1; m >>= 1) vv += __shfl_xor(vv, m);
    const float inv = rsqrtf(vv * (1.0f / 256.0f) + 1e-5f);
    float* orow = out + (size_t)b * 256;
#pragma unroll
    for (int c = lane; c < 256; c += 32) {
      const float val = (row[c] - mu) * inv * g2[c] + be2[c];
      orow[c] = val > 0.f ? val : 0.f;
    }
  }
}

extern "C" void kernel_launch(void* const* d_in, const int* in_sizes, int n_in,
                              void* d_out, int out_size, void* d_ws,
                              size_t ws_size, hipStream_t stream) {
  const float* x = (const float*)d_in[0];
  const float* w_in = (const float*)d_in[1];
  const float* b_in = (const float*)d_in[2];
  const float* w0 = (const float*)d_in[3];
  const float* as0 = (const float*)d_in[4];
  const float* ad0 = (const float*)d_in[5];
  const float* bias0 = (const float*)d_in[6];
  const float* w1 = (const float*)d_in[7];
  const float* as1 = (const float*)d_in[8];
  const float* ad1 = (const float*)d_in[9];
  const float* bias1 = (const float*)d_in[10];
  const float* w2 = (const float*)d_in[11];
  const float* as2 = (const float*)d_in[12];
  const float* ad2 = (const float*)d_in[13];
  const float* bias2 = (const float*)d_in[14];
  const float* mw1 = (const float*)d_in[15];
  const float* mb1 = (const float*)d_in[16];
  const float* g1 = (const float*)d_in[17];
  const float* be1 = (const float*)d_in[18];
  const float* mw2 = (const float*)d_in[19];
  const float* mb2 = (const float*)d_in[20];
  const float* g2 = (const float*)d_in[21];
  const float* be2 = (const float*)d_in[22];
  float* out = (float*)d_out;

  const int B = in_sizes[0] / 256;   // 8192
  const int blocks = B / 8;          // 8 batch elements per workgroup
  const size_t lds_bytes = (size_t)(8 * 9344 + 16 * 64) * sizeof(float);

  gat_fused<<<blocks, 256, lds_bytes, stream>>>(
      x, w_in, b_in, w0, as0, ad0, bias0, w1, as1, ad1, bias1, w2, as2, ad2,
      bias2, mw1, mb1, g1, be1, mw2, mb2, g2, be2, out);
}